// Cluster_42889543417975
// MI455X (gfx1250) — compile-verified
//
#include <hip/hip_runtime.h>

typedef __attribute__((ext_vector_type(2))) float v2f;
typedef __attribute__((ext_vector_type(4))) float v4f;
typedef __attribute__((ext_vector_type(8))) float v8f;

#define NCH   96      // HDIM == CV == 96
#define KX    32      // DIM
#define KS    31      // DIM-1
#define HW    512
#define TPB   256
#define NWAVE 8

#define CF_SIZE  (2*256*96)
#define LAB_OFF  CF_SIZE
#define LAB_SIZE (2*16*256)
#define SPIX_OFF (CF_SIZE + LAB_SIZE)

struct __align__(16) Smem {
  float WfS[NCH*KX];      // [96][32]
  float WsdfS[NCH*KX];    // [96][32] (col 31 zero-padded)
  float bfS[NCH];
  float bsdfS[NCH];
  float cenPF[NCH*4];     // pooled pf means (value centers / iter-0 pf centers)
  float cenDF[NCH*4];     // conv centers via linearity
  float cenSF[NCH*4];
  float cenPF1[NCH*4];    // updated pf centers after iter 0
  float poolX[KX*4];
  float poolSD[KX*4];
  float numW[NWAVE][NCH*4];
  float denW[NWAVE][4];
  float labC[64];         // [16 cls][4 s]
  float csqDS[4];         // sum of squared df+sf centers per s
  float csqPF0[4];
  float csqPF1[4];
};

__device__ __forceinline__ v8f wmma4(v2f a, v2f b, v8f c) {
  // D(16x16,f32) = A(16x4,f32) x B(4x16,f32) + C
  return __builtin_amdgcn_wmma_f32_16x16x4_f32(false, a, false, b, (short)0, c,
                                               false, false);
}

__global__ __launch_bounds__(TPB) void ssn_kernel(
    const float* __restrict__ x, const float* __restrict__ feat,
    const float* __restrict__ sdfd, const float* __restrict__ Wf,
    const float* __restrict__ bf, const float* __restrict__ Wsdf,
    const float* __restrict__ bsdf, const int* __restrict__ gt,
    float* __restrict__ out)
{
  __shared__ Smem sm;
  const int tid   = threadIdx.x;
  const int lane  = tid & 31;
  const int wv    = tid >> 5;
  const int nlane = lane & 15;   // pixel / column lane
  const int klane = lane >> 4;   // half select

  const int t  = blockIdx.x;       // 128 tiles
  const int Bi = t >> 6;
  const int f1 = (t >> 3) & 7;
  const int f2 = t & 7;
  const int tl = f1*8 + f2;
  const int y0 = f1*64, x0 = f2*64;

  const float* xB = x    + (size_t)Bi*KX *HW*HW;
  const float* fB = feat + (size_t)Bi*NCH*HW*HW;
  const float* sB = sdfd + (size_t)Bi*KS *HW*HW;

  // ---- stage weights / zero accumulators ----
  for (int i=tid;i<NCH*KX;i+=TPB) sm.WfS[i] = Wf[i];
  for (int i=tid;i<NCH*KX;i+=TPB){ int c=i>>5,k=i&31; sm.WsdfS[i]=(k<KS)?Wsdf[c*KS+k]:0.f; }
  for (int i=tid;i<NCH;i+=TPB){ sm.bfS[i]=bf[i]; sm.bsdfS[i]=bsdf[i]; }
  for (int i=tid;i<KX*4;i+=TPB){ sm.poolX[i]=0.f; sm.poolSD[i]=0.f; }
  for (int i=tid;i<64;i+=TPB) sm.labC[i]=0.f;
  __syncthreads();

  // ---- pass A: pooling (feat 96 ch, x 32 ch, sdf 31 ch) x 4 quadrants ----
  #pragma unroll 1
  for (int task = wv; task < 159*4; task += NWAVE) {
    int q = task & 3, m = task >> 2;
    int qy = q >> 1, qx = q & 1;
    const float* src; int ch;
    if (m < 96)        { src = fB; ch = m; }
    else if (m < 128)  { src = xB; ch = m-96; }
    else               { src = sB; ch = m-128; }
    const float* rp = src + ((size_t)ch*HW + (y0+qy*32))*HW + (x0+qx*32) + lane;
    float s = 0.f;
    #pragma unroll 8
    for (int rr=0; rr<32; ++rr) s += rp[rr*HW];
    for (int off=16; off; off>>=1) s += __shfl_xor(s, off, 32);
    if (lane==0) {
      float mv = s * (1.0f/1024.0f);
      if (m<96)       sm.cenPF[m*4+q]        = mv;
      else if (m<128) sm.poolX[(m-96)*4+q]   = mv;
      else            sm.poolSD[(m-128)*4+q] = mv;
    }
  }
  __syncthreads();
  // conv centers via linearity: mean(W*x+b) = W*mean(x)+b
  #pragma unroll 1
  for (int i=tid; i<NCH*4*2; i+=TPB) {
    int which = (i >= NCH*4);
    int j = i - which*NCH*4;
    int c = j >> 2, q = j & 3;
    const float* Wm = which ? sm.WsdfS : sm.WfS;
    const float* pm = which ? sm.poolSD : sm.poolX;
    float acc = which ? sm.bsdfS[c] : sm.bfS[c];
    #pragma unroll 8
    for (int k=0;k<KX;k++) acc += Wm[c*KX+k]*pm[k*4+q];
    (which ? sm.cenSF : sm.cenDF)[j] = acc;
  }
  __syncthreads();
  if (tid < 4) {
    float a=0.f, b2=0.f;
    #pragma unroll 4
    for (int c=0;c<NCH;c++){
      float d1=sm.cenDF[c*4+tid], d2=sm.cenSF[c*4+tid], p=sm.cenPF[c*4+tid];
      a += d1*d1 + d2*d2; b2 += p*p;
    }
    sm.csqDS[tid]=a; sm.csqPF0[tid]=b2;
  }
  __syncthreads();

  // ---- passes 0/1: SSN iterations ----
  #pragma unroll 1
  for (int pass=0; pass<2; ++pass) {
    const float* cenP = pass ? sm.cenPF1 : sm.cenPF;
    const float* csqP = pass ? sm.csqPF1 : sm.csqPF0;
    v8f Cacc[6];
    #pragma unroll
    for (int o=0;o<6;o++) for (int r=0;r<8;r++) Cacc[o][r]=0.f;
    float dn0=0.f,dn1=0.f,dn2=0.f,dn3=0.f;

    #pragma unroll 1
    for (int it=0; it<32; ++it) {
      int n0 = (wv*32 + it) * 16;        // 16 consecutive pixels in one row
      int gy = y0 + (n0 >> 6);
      int gx = x0 + (n0 & 63);

      // conv (WMMA) + dot with conv centers, for df and sf
      float dD0=0,dD1=0,dD2=0,dD3=0, sqDS=0;
      #pragma unroll 1
      for (int mat=0; mat<2; ++mat) {
        const float* Wl   = mat ? sm.WsdfS : sm.WfS;
        const float* bl   = mat ? sm.bsdfS : sm.bfS;
        const float* cl   = mat ? sm.cenSF : sm.cenDF;
        const float* gsrc = mat ? sB : xB;
        const int KM      = mat ? KS : KX;
        #pragma unroll 1
        for (int ocb=0; ocb<6; ++ocb) {
          v8f acc;
          #pragma unroll
          for (int r=0;r<8;r++) acc[r]=0.f;
          #pragma unroll
          for (int ks=0; ks<8; ++ks) {
            v2f a = *(const v2f*)&Wl[(ocb*16 + nlane)*KX + ks*4 + klane*2];
            int k0 = ks*4 + klane*2;
            v2f b;
            b.x = (k0   < KM) ? gsrc[((size_t)k0    *HW + gy)*HW + gx + nlane] : 0.f;
            b.y = (k0+1 < KM) ? gsrc[((size_t)(k0+1)*HW + gy)*HW + gx + nlane] : 0.f;
            acc = wmma4(a, b, acc);
          }
          #pragma unroll
          for (int r=0;r<8;r++) {
            int c = ocb*16 + klane*8 + r;
            float v = acc[r] + bl[c];
            v4f cen = *(const v4f*)&cl[c*4];
            dD0 += v*cen.x; dD1 += v*cen.y; dD2 += v*cen.z; dD3 += v*cen.w;
            sqDS += v*v;
          }
        }
      }
      // pf dot with pf centers (lane = pixel, halves split 96 channels)
      float dP0=0,dP1=0,dP2=0,dP3=0, sqPF=0;
      {
        const int c0 = klane*48;
        const float* pp = fB + ((size_t)c0*HW + gy)*HW + gx + nlane;
        #pragma unroll 4
        for (int cc=0; cc<48; ++cc) {
          float v = pp[(size_t)cc*HW*HW];
          v4f cen = *(const v4f*)&cenP[(c0+cc)*4];
          dP0 += v*cen.x; dP1 += v*cen.y; dP2 += v*cen.z; dP3 += v*cen.w;
          sqPF += v*v;
        }
      }
      dD0 += __shfl_xor(dD0,16,32); dD1 += __shfl_xor(dD1,16,32);
      dD2 += __shfl_xor(dD2,16,32); dD3 += __shfl_xor(dD3,16,32);
      dP0 += __shfl_xor(dP0,16,32); dP1 += __shfl_xor(dP1,16,32);
      dP2 += __shfl_xor(dP2,16,32); dP3 += __shfl_xor(dP3,16,32);
      sqDS += __shfl_xor(sqDS,16,32); sqPF += __shfl_xor(sqPF,16,32);

      float d0 = sqDS + sm.csqDS[0] - 2.f*dD0 + sqPF + csqP[0] - 2.f*dP0;
      float d1 = sqDS + sm.csqDS[1] - 2.f*dD1 + sqPF + csqP[1] - 2.f*dP1;
      float d2 = sqDS + sm.csqDS[2] - 2.f*dD2 + sqPF + csqP[2] - 2.f*dP2;
      float d3 = sqDS + sm.csqDS[3] - 2.f*dD3 + sqPF + csqP[3] - 2.f*dP3;

      float mn  = fminf(fminf(d0,d1),fminf(d2,d3));
      float e0=expf(mn-d0), e1=expf(mn-d1), e2=expf(mn-d2), e3=expf(mn-d3);
      float inv = 1.f/(e0+e1+e2+e3);
      float w0,w1,w2,w3;
      if (pass==0) {
        w0=e0*inv; w1=e1*inv; w2=e2*inv; w3=e3*inv;             // soft A0
      } else {
        int sb=0; float bd=d0;
        if (d1<bd){bd=d1;sb=1;}
        if (d2<bd){bd=d2;sb=2;}
        if (d3<bd){bd=d3;sb=3;}
        float es   = (sb==0)?e0:(sb==1)?e1:(sb==2)?e2:e3;
        float simv = es*inv;
        w0=(sb==0)?simv:0.f; w1=(sb==1)?simv:0.f;
        w2=(sb==2)?simv:0.f; w3=(sb==3)?simv:0.f;               // masked sim
        if (klane==0) {
          size_t pidx = ((size_t)Bi*HW + gy)*HW + gx + nlane;
          out[SPIX_OFF + pidx] = (float)(tl*4 + sb);
          int cls = gt[pidx];
          atomicAdd(&sm.labC[cls*4 + sb], 1.0f);                // exact integer adds
        }
      }
      if (klane==0) { dn0+=w0; dn1+=w1; dn2+=w2; dn3+=w3; }

      // accumulate num[c,s] += w[s,n]*pf[c,n] via WMMA (A=pf frag, B=weights)
      #pragma unroll 1
      for (int ksp=0; ksp<4; ++ksp) {
        v2f bb;
        #pragma unroll
        for (int j=0;j<2;++j) {
          int q = ksp*4 + klane*2 + j;
          float a0=__shfl(w0,q,32), a1=__shfl(w1,q,32);
          float a2=__shfl(w2,q,32), a3=__shfl(w3,q,32);
          float bv = (nlane==0)?a0:(nlane==1)?a1:(nlane==2)?a2:(nlane==3)?a3:0.f;
          if (j==0) bb.x=bv; else bb.y=bv;
        }
        #pragma unroll
        for (int ocb=0; ocb<6; ++ocb) {
          v2f a = *(const v2f*)(fB + ((size_t)(ocb*16+nlane)*HW + gy)*HW
                                   + gx + ksp*4 + klane*2);
          Cacc[ocb] = wmma4(a, bb, Cacc[ocb]);
        }
      }
    } // chunks

    // deterministic cross-wave reduction via per-wave LDS slots
    if (nlane < 4) {
      #pragma unroll 1
      for (int ocb=0; ocb<6; ++ocb)
        for (int r=0;r<8;r++)
          sm.numW[wv][(ocb*16 + klane*8 + r)*4 + nlane] = Cacc[ocb][r];
    }
    {
      float v0=dn0,v1=dn1,v2=dn2,v3=dn3;
      #pragma unroll
      for (int off=16; off; off>>=1) {
        v0 += __shfl_xor(v0,off,32); v1 += __shfl_xor(v1,off,32);
        v2 += __shfl_xor(v2,off,32); v3 += __shfl_xor(v3,off,32);
      }
      if (lane==0){ sm.denW[wv][0]=v0; sm.denW[wv][1]=v1;
                    sm.denW[wv][2]=v2; sm.denW[wv][3]=v3; }
    }
    __syncthreads();
    if (pass==0) {
      #pragma unroll 1
      for (int i=tid;i<NCH*4;i+=TPB){
        int q=i&3;
        float ns=0.f, ds=0.f;
        #pragma unroll
        for (int w=0;w<NWAVE;w++){ ns+=sm.numW[w][i]; ds+=sm.denW[w][q]; }
        sm.cenPF1[i] = ns/(ds+1e-16f);
      }
      __syncthreads();
      if (tid<4){
        float a=0.f;
        #pragma unroll 4
        for (int c=0;c<NCH;c++){ float p=sm.cenPF1[c*4+tid]; a+=p*p; }
        sm.csqPF1[tid]=a;
      }
      __syncthreads();
    }
  } // passes

  // ---- final outputs ----
  #pragma unroll 1
  for (int i=tid;i<NCH*4;i+=TPB){
    int c=i>>2, s=i&3;
    float ns=0.f, ds=0.f;
    #pragma unroll
    for (int w=0;w<NWAVE;w++){ ns+=sm.numW[w][i]; ds+=sm.denW[w][s]; }
    float val = (ns + sm.cenPF[i]) / (ds + 1.0f);
    out[((size_t)Bi*256 + tl*4 + s)*NCH + c] = val;   // center_feat
  }
  #pragma unroll 1
  for (int i=tid;i<64;i+=TPB){
    int cls=i>>2, s=i&3;
    out[LAB_OFF + ((size_t)Bi*16 + cls)*256 + tl*4 + s] = sm.labC[i];
  }
}

extern "C" void kernel_launch(void* const* d_in, const int* in_sizes, int n_in,
                              void* d_out, int out_size, void* d_ws, size_t ws_size,
                              hipStream_t stream) {
  (void)in_sizes; (void)n_in; (void)d_ws; (void)ws_size; (void)out_size;
  const float* x    = (const float*)d_in[0];
  const float* feat = (const float*)d_in[1];
  const float* sdfd = (const float*)d_in[2];
  const float* Wf   = (const float*)d_in[3];
  const float* bf   = (const float*)d_in[4];
  const float* Wsdf = (const float*)d_in[5];
  const float* bsdf = (const float*)d_in[6];
  const int*   gt   = (const int*)d_in[7];
  float* out = (float*)d_out;
  ssn_kernel<<<dim3(128), dim3(256), 0, stream>>>(x, feat, sdfd, Wf, bf, Wsdf,
                                                  bsdf, gt, out);
}